// IndexStaged_35519379538453
// MI455X (gfx1250) — compile-verified
//
#include <hip/hip_runtime.h>
#include <hip/hip_bf16.h>

#define N_TOT   200000
#define NSUB    50000
#define D_FULL  128
#define DP      64
#define DEG     32
#define NENTRY  32
#define BQ      128
#define KOUT    10
#define EF      64
#define NCAND   (EF + DEG)   // 96
#define ITERS   24
#define INF_F   1e30f

typedef __attribute__((ext_vector_type(2))) float v2f;
typedef __attribute__((ext_vector_type(8))) float v8f;

// ---------------------------------------------------------------------------
// Kernel 1: qf = query @ VT^T  (128x128x128, f32) via V_WMMA_F32_16X16X4_F32.
// One wave (32 lanes) per 16x16 output tile; K accumulated in steps of 4.
// ---------------------------------------------------------------------------
__global__ __launch_bounds__(32)
void rot_gemm_wmma(const float* __restrict__ query,
                   const float* __restrict__ VT,
                   float* __restrict__ qf) {
    const int n0   = blockIdx.x * 16;      // output feature tile (rows of VT)
    const int m0   = blockIdx.y * 16;      // query tile
    const int lane = threadIdx.x;          // wave32
    const int lm   = lane & 15;            // M (for A) / N (for B,D) index
    const int half = lane >> 4;            // K-half selector

    v8f c = {};
    #pragma unroll
    for (int k0 = 0; k0 < D_FULL; k0 += 4) {
        const int k = k0 + 2 * half;
        // A fragment: 16x4 tile of query.  v0: K={0,2}, v1: K={1,3} per half.
        v2f a;
        a.x = query[(m0 + lm) * D_FULL + k + 0];
        a.y = query[(m0 + lm) * D_FULL + k + 1];
        // B fragment: 4x16 tile of VT^T, i.e. B[k][n] = VT[n0+n][k0+k].
        v2f b;
        b.x = VT[(n0 + lm) * D_FULL + k + 0];
        b.y = VT[(n0 + lm) * D_FULL + k + 1];
        c = __builtin_amdgcn_wmma_f32_16x16x4_f32(
                /*neg_a=*/false, a, /*neg_b=*/false, b,
                /*c_mod=*/(short)0, c, /*reuse_a=*/false, /*reuse_b=*/false);
    }
    // D layout: VGPR r -> row m0+r (lanes 0-15) / m0+r+8 (lanes 16-31), col n0+lm.
    #pragma unroll
    for (int r = 0; r < 8; ++r) {
        qf[(m0 + r + 8 * half) * D_FULL + (n0 + lm)] = c[r];
    }
}

// ---------------------------------------------------------------------------
// Kernel 2: two-stage beam search. One 128-thread block per query.
// Latency-bound random gather: state in LDS, b128 loads, prefetch hints.
// ---------------------------------------------------------------------------
__device__ __forceinline__
float l2sq_row(const float* __restrict__ row, const float* __restrict__ sq, int dim) {
    float acc = 0.0f;
    #pragma unroll 4
    for (int d = 0; d < dim; d += 4) {
        float4 v = *(const float4*)(row + d);
        float d0 = sq[d + 0] - v.x;
        float d1 = sq[d + 1] - v.y;
        float d2 = sq[d + 2] - v.z;
        float d3 = sq[d + 3] - v.w;
        acc += d0 * d0 + d1 * d1 + d2 * d2 + d3 * d3;
    }
    return acc;
}

__device__ __forceinline__
void beam_stage(const int* __restrict__ nbrs_mat,
                const float* __restrict__ storage, int dim,
                float* sq, int* cI, float* cD, int* cE,
                int* tI, float* tD, int* tE, int* sBid, int t) {
    for (int it = 0; it < ITERS; ++it) {
        // argmin over unexpanded candidates (matches jnp.argmin(where(ex,INF,cD)))
        if (t == 0) {
            float bv = INF_F;
            int   bi = 0;
            for (int j = 0; j < EF; ++j) {
                if (!cE[j] && cD[j] < bv) { bv = cD[j]; bi = j; }
            }
            cE[bi] = 1;
            int id = cI[bi];
            *sBid = (id < 0) ? 0 : id;   // jnp.clip(cI[best], 0)
        }
        __syncthreads();
        const int bid = *sBid;

        if (t < DEG) {
            const int nid = nbrs_mat[bid * DEG + t];
            const float* row = storage + (long)nid * dim;
            __builtin_prefetch(row, 0, 3);           // global_prefetch_b8
            float acc = l2sq_row(row, sq, dim);
            bool dup = false;
            for (int j = 0; j < EF; ++j) dup |= (cI[j] == nid);
            cI[EF + t] = nid;
            cD[EF + t] = dup ? INF_F : acc;
            cE[EF + t] = 0;
        }
        __syncthreads();

        // keep EF smallest of NCAND; rank == top_k(-d) order (index tie-break)
        if (t < NCAND) {
            const float di = cD[t];
            int rank = 0;
            for (int j = 0; j < NCAND; ++j) {
                const float dj = cD[j];
                rank += (dj < di) || (dj == di && j < t);
            }
            if (rank < EF) { tI[rank] = cI[t]; tD[rank] = di; tE[rank] = cE[t]; }
        }
        __syncthreads();
        if (t < EF) { cI[t] = tI[t]; cD[t] = tD[t]; cE[t] = tE[t]; }
        __syncthreads();
    }
}

__global__ __launch_bounds__(128)
void beam_search_kernel(const float* __restrict__ qf,
                        const float* __restrict__ storage,
                        const float* __restrict__ storage_cuda,
                        const float* __restrict__ entry_vectors,
                        const int*   __restrict__ nodelist,
                        const int*   __restrict__ full_nbrs,
                        const int*   __restrict__ sub_nbrs,
                        const int*   __restrict__ entry_nodes,
                        float* __restrict__ out) {
    __shared__ float sq[D_FULL];
    __shared__ int   cI[NCAND];
    __shared__ float cD[NCAND];
    __shared__ int   cE[NCAND];
    __shared__ int   tI[EF];
    __shared__ float tD[EF];
    __shared__ int   tE[EF];
    __shared__ int   sBid;

    const int b = blockIdx.x;
    const int t = threadIdx.x;

    // query row (rotated); qp = sq[0:64]
    if (t < D_FULL) sq[t] = qf[b * D_FULL + t];
    __syncthreads();

    // ---- stage 1 init: 32 entry nodes + 32 INF padding slots ----
    if (t < NENTRY) {
        float acc = 0.0f;
        const float* ev = entry_vectors + t * DP;
        #pragma unroll 4
        for (int d = 0; d < DP; d += 4) {
            float4 v = *(const float4*)(ev + d);
            float d0 = sq[d + 0] - v.x;
            float d1 = sq[d + 1] - v.y;
            float d2 = sq[d + 2] - v.z;
            float d3 = sq[d + 3] - v.w;
            acc += d0 * d0 + d1 * d1 + d2 * d2 + d3 * d3;
        }
        cI[t] = entry_nodes[t];
        cD[t] = acc;
        cE[t] = 0;
    } else if (t < EF) {
        cI[t] = -1;
        cD[t] = INF_F;
        cE[t] = 1;
    }
    __syncthreads();

    // ---- stage 1: subgraph beam (dim 64) ----
    beam_stage(sub_nbrs, storage_cuda, DP, sq, cI, cD, cE, tI, tD, tE, &sBid, t);

    // ---- transition: map to full ids, recompute full-dim distances ----
    if (t < EF) {
        const int   i1 = cI[t];
        const float d1 = cD[t];
        const int   ci = (i1 < 0) ? 0 : i1;
        const int  fid = nodelist[ci];
        const float* row = storage + (long)fid * D_FULL;
        float acc = l2sq_row(row, sq, D_FULL);
        cI[t] = fid;
        cD[t] = (d1 >= 0.5f * INF_F) ? INF_F : acc;
        cE[t] = 0;
    }
    __syncthreads();

    // ---- stage 2: full-graph beam (dim 128) ----
    beam_stage(full_nbrs, storage, D_FULL, sq, cI, cD, cE, tI, tD, tE, &sBid, t);

    // ---- final top-K: smallest KOUT of EF, sorted ascending ----
    if (t < EF) {
        const float di = cD[t];
        int rank = 0;
        for (int j = 0; j < EF; ++j) {
            const float dj = cD[j];
            rank += (dj < di) || (dj == di && j < t);
        }
        if (rank < KOUT) {
            out[b * KOUT + rank]             = (float)cI[t];   // indices block
            out[BQ * KOUT + b * KOUT + rank] = di;             // distances block
        }
    }
}

// ---------------------------------------------------------------------------
extern "C" void kernel_launch(void* const* d_in, const int* in_sizes, int n_in,
                              void* d_out, int out_size, void* d_ws, size_t ws_size,
                              hipStream_t stream) {
    const float* query         = (const float*)d_in[0];
    const float* VT            = (const float*)d_in[1];
    const float* storage       = (const float*)d_in[2];
    const float* storage_cuda  = (const float*)d_in[3];
    const float* entry_vectors = (const float*)d_in[4];
    const int*   nodelist      = (const int*)d_in[5];
    const int*   full_nbrs     = (const int*)d_in[6];
    const int*   sub_nbrs      = (const int*)d_in[7];
    const int*   entry_nodes   = (const int*)d_in[8];
    (void)in_sizes; (void)n_in; (void)out_size; (void)ws_size;

    float* qf  = (float*)d_ws;           // 128*128 floats = 64 KB scratch
    float* out = (float*)d_out;

    // 1) rotation GEMM on WMMA: 8x8 tiles of 16x16, one wave each
    dim3 ggrid(D_FULL / 16, BQ / 16);
    rot_gemm_wmma<<<ggrid, 32, 0, stream>>>(query, VT, qf);

    // 2) beam search: one 128-thread workgroup per query
    beam_search_kernel<<<BQ, 128, 0, stream>>>(qf, storage, storage_cuda,
                                               entry_vectors, nodelist,
                                               full_nbrs, sub_nbrs, entry_nodes,
                                               out);
}